// GNN_11587821764812
// MI455X (gfx1250) — compile-verified
//
#include <hip/hip_runtime.h>
#include <hip/hip_bf16.h>

// GCN 2-layer forward for MI455X (gfx1250, wave32).
// Dense GEMM h1 = x @ W1 uses V_WMMA_F32_16X16X4_F32 (full fp32 precision,
// N=16 == hidden dim matches WMMA tile). Edge aggregation is L2-resident
// atomic scatter-add (h1/agg1 = 12.8 MB each << 192 MB L2).

typedef __attribute__((ext_vector_type(2))) float v2f;
typedef __attribute__((ext_vector_type(8))) float v8f;

#define F_IN 128
#define F_H  16
#define F_C  2

// ---------------- utility ----------------
__global__ void GNN_zero_f32(float* __restrict__ p, long n) {
    long i = (long)blockIdx.x * blockDim.x + threadIdx.x;
    if (i < n) p[i] = 0.0f;
}

// deg[dst] += 1 for every edge
__global__ void GNN_degree(const int* __restrict__ eidx, long E,
                           float* __restrict__ deg) {
    long e = (long)blockIdx.x * blockDim.x + threadIdx.x;
    if (e >= E) return;
    int d = eidx[E + e];
    atomicAdd(&deg[d], 1.0f);
}

// dinv = rsqrt(deg + 1)   (+1 = self loop)
__global__ void GNN_dinv(const float* __restrict__ deg,
                         float* __restrict__ dinv, int n) {
    int i = blockIdx.x * blockDim.x + threadIdx.x;
    if (i < n) dinv[i] = rsqrtf(deg[i] + 1.0f);
}

// ---------------- h1 = x @ W1 via V_WMMA_F32_16X16X4_F32 ----------------
// One wave handles one 16-row tile of x; K=128 swept 4 at a time.
// A (16x4 f32): lanes 0-15 -> {K=0,K=1}, lanes 16-31 -> {K=2,K=3}  (M = lane&15)
// B (4x16 f32): lanes 0-15 -> rows K=0,1; lanes 16-31 -> rows K=2,3 (N = lane&15)
// C/D (16x16 f32): VGPR r: lanes 0-15 M=r, lanes 16-31 M=r+8; N = lane&15
__global__ void GNN_gemm1_wmma(const float* __restrict__ x,
                               const float* __restrict__ W1,
                               float* __restrict__ h1, int nTiles) {
    int lane = threadIdx.x & 31;
    int wave = threadIdx.x >> 5;
    int tile = blockIdx.x * (blockDim.x >> 5) + wave;
    if (tile >= nTiles) return;              // wave-uniform: EXEC stays all-1s

    int m    = lane & 15;                    // row of A / col of B for this lane
    int koff = (lane >> 4) << 1;             // 0 for lanes 0-15, 2 for lanes 16-31
    long rowA = (long)(tile * 16 + m) * F_IN;

    v8f acc = {};
    #pragma unroll 4
    for (int kk = 0; kk < F_IN; kk += 4) {
        v2f a, b;
        a.x = x[rowA + kk + koff];
        a.y = x[rowA + kk + koff + 1];
        b.x = W1[(kk + koff)     * F_H + m];
        b.y = W1[(kk + koff + 1) * F_H + m];
        // 8 args: (neg_a, A, neg_b, B, c_mod, C, reuse_a, reuse_b)
        acc = __builtin_amdgcn_wmma_f32_16x16x4_f32(
            false, a, false, b, (short)0, acc, false, false);
    }

    int rbase = tile * 16 + (lane >> 4) * 8;
    #pragma unroll
    for (int r = 0; r < 8; ++r)
        h1[(long)(rbase + r) * F_H + m] = acc[r];
}

// scalar tail in case node count is not a multiple of 16 (not hit for N=200000)
__global__ void GNN_gemm1_tail(const float* __restrict__ x,
                               const float* __restrict__ W1,
                               float* __restrict__ h1, int startRow, int n) {
    int idx = blockIdx.x * blockDim.x + threadIdx.x;
    int row = startRow + (idx >> 4);
    int col = idx & 15;
    if (row >= n) return;
    float acc = 0.0f;
    for (int k = 0; k < F_IN; ++k)
        acc += x[(long)row * F_IN + k] * W1[k * F_H + col];
    h1[(long)row * F_H + col] = acc;
}

// ---------------- edge scatter, layer 1: agg1[dst] += norm * h1[src] --------
__global__ void GNN_scatter1(const int* __restrict__ eidx, long E,
                             const float* __restrict__ dinv,
                             const float* __restrict__ h1,
                             float* __restrict__ agg1) {
    long e = (long)blockIdx.x * blockDim.x + threadIdx.x;
    if (e >= E) return;
    int s = eidx[e];
    int d = eidx[E + e];
    float norm = dinv[s] * dinv[d];
    const float4* hp = (const float4*)(h1 + (long)s * F_H);  // L2-resident gather
    float* ap = agg1 + (long)d * F_H;
    #pragma unroll
    for (int q = 0; q < 4; ++q) {
        float4 v = hp[q];
        atomicAdd(ap + q * 4 + 0, norm * v.x);
        atomicAdd(ap + q * 4 + 1, norm * v.y);
        atomicAdd(ap + q * 4 + 2, norm * v.z);
        atomicAdd(ap + q * 4 + 3, norm * v.w);
    }
}

// ---- combine layer1 (+bias, ReLU) fused with tiny 16x2 GEMM -> h2 ----------
__global__ void GNN_combine1(const float* __restrict__ agg1,
                             const float* __restrict__ h1,
                             const float* __restrict__ dinv,
                             const float* __restrict__ b1,
                             const float* __restrict__ W2,
                             float* __restrict__ h2, int n) {
    int i = blockIdx.x * blockDim.x + threadIdx.x;
    if (i >= n) return;
    float di  = dinv[i];
    float di2 = di * di;
    float acc0 = 0.0f, acc1 = 0.0f;
    #pragma unroll
    for (int j = 0; j < F_H; ++j) {
        float v = agg1[(long)i * F_H + j] + di2 * h1[(long)i * F_H + j] + b1[j];
        v = fmaxf(v, 0.0f);                        // ReLU
        acc0 = fmaf(v, W2[j * F_C + 0], acc0);
        acc1 = fmaf(v, W2[j * F_C + 1], acc1);
    }
    h2[(long)i * F_C + 0] = acc0;
    h2[(long)i * F_C + 1] = acc1;
}

// ---------------- edge scatter, layer 2 -------------------------------------
__global__ void GNN_scatter2(const int* __restrict__ eidx, long E,
                             const float* __restrict__ dinv,
                             const float* __restrict__ h2,
                             float* __restrict__ agg2) {
    long e = (long)blockIdx.x * blockDim.x + threadIdx.x;
    if (e >= E) return;
    int s = eidx[e];
    int d = eidx[E + e];
    float norm = dinv[s] * dinv[d];
    float2 v = *(const float2*)(h2 + (long)s * F_C);
    atomicAdd(&agg2[(long)d * F_C + 0], norm * v.x);
    atomicAdd(&agg2[(long)d * F_C + 1], norm * v.y);
}

// ---------------- final: self term + bias + log_softmax over 2 classes ------
__global__ void GNN_final(const float* __restrict__ agg2,
                          const float* __restrict__ h2,
                          const float* __restrict__ dinv,
                          const float* __restrict__ b2,
                          float* __restrict__ out, int n) {
    int i = blockIdx.x * blockDim.x + threadIdx.x;
    if (i >= n) return;
    float di  = dinv[i];
    float di2 = di * di;
    float o0 = agg2[(long)i * F_C + 0] + di2 * h2[(long)i * F_C + 0] + b2[0];
    float o1 = agg2[(long)i * F_C + 1] + di2 * h2[(long)i * F_C + 1] + b2[1];
    float mx  = fmaxf(o0, o1);
    float lse = mx + logf(expf(o0 - mx) + expf(o1 - mx));
    out[(long)i * F_C + 0] = o0 - lse;
    out[(long)i * F_C + 1] = o1 - lse;
}

// ---------------------------------------------------------------------------
extern "C" void kernel_launch(void* const* d_in, const int* in_sizes, int n_in,
                              void* d_out, int out_size, void* d_ws, size_t ws_size,
                              hipStream_t stream) {
    const float* x  = (const float*)d_in[0];
    const int* eidx = (const int*)  d_in[1];   // [2, E] row-major: src then dst
    const float* W1 = (const float*)d_in[2];   // [128,16]
    const float* b1 = (const float*)d_in[3];   // [16]
    const float* W2 = (const float*)d_in[4];   // [16,2]
    const float* b2 = (const float*)d_in[5];   // [2]
    float* out = (float*)d_out;                // [N,2]

    long nN = (long)in_sizes[0] / F_IN;        // 200000
    long E  = (long)in_sizes[1] / 2;           // 6400000

    // Workspace layout (floats). Zero-init regions are contiguous at offset 0.
    float* ws   = (float*)d_ws;
    float* deg  = ws;                  // [nN]      (zeroed)
    float* agg1 = ws + nN;             // [nN*16]   (zeroed)
    float* agg2 = ws + 17 * nN;        // [nN*2]    (zeroed)
    float* dinv = ws + 19 * nN;        // [nN]
    float* h1   = ws + 20 * nN;        // [nN*16]
    float* h2   = ws + 36 * nN;        // [nN*2]   -- total 38*nN floats (~30.4 MB)

    const int T = 256;
    long zn = 19 * nN;
    GNN_zero_f32<<<(int)((zn + T - 1) / T), T, 0, stream>>>(ws, zn);
    GNN_degree  <<<(int)((E + T - 1) / T), T, 0, stream>>>(eidx, E, deg);
    GNN_dinv    <<<(int)((nN + T - 1) / T), T, 0, stream>>>(deg, dinv, (int)nN);

    int nTiles = (int)(nN / 16);                       // 12500 full WMMA tiles
    if (nTiles > 0)
        GNN_gemm1_wmma<<<(nTiles + 7) / 8, T, 0, stream>>>(x, W1, h1, nTiles);
    int rem = (int)(nN - (long)nTiles * 16);
    if (rem > 0)
        GNN_gemm1_tail<<<(rem * 16 + T - 1) / T, T, 0, stream>>>(
            x, W1, h1, nTiles * 16, (int)nN);

    GNN_scatter1<<<(int)((E + T - 1) / T), T, 0, stream>>>(eidx, E, dinv, h1, agg1);
    GNN_combine1<<<(int)((nN + T - 1) / T), T, 0, stream>>>(agg1, h1, dinv, b1, W2, h2, (int)nN);
    GNN_scatter2<<<(int)((E + T - 1) / T), T, 0, stream>>>(eidx, E, dinv, h2, agg2);
    GNN_final   <<<(int)((nN + T - 1) / T), T, 0, stream>>>(agg2, h2, dinv, b2, out, (int)nN);
}